// ProposalModule_pred_55705725829193
// MI455X (gfx1250) — compile-verified
//
#include <hip/hip_runtime.h>
#include <hip/hip_bf16.h>

typedef __attribute__((ext_vector_type(2))) float v2f;
typedef __attribute__((ext_vector_type(8))) float v8f;
typedef __attribute__((ext_vector_type(4))) unsigned int v4u;
typedef __attribute__((ext_vector_type(8))) int v8i;
typedef __attribute__((ext_vector_type(4))) int v4i;

#define B_    256
#define K_    256
#define C_    128
#define C2_   256
#define KG_   64
#define O3_   117
#define OUTF_ 183
#define PI_OVER_12 0.26179938779914946f

#if defined(__has_builtin)
#if __has_builtin(__builtin_amdgcn_tensor_load_to_lds) && __has_builtin(__builtin_amdgcn_s_wait_tensorcnt)
#define HAVE_TDM 1
#endif
#endif

__device__ __forceinline__ v8f wmma4(v2f a, v2f b, v8f c) {
  return __builtin_amdgcn_wmma_f32_16x16x4_f32(false, a, false, b, (short)0, c,
                                               false, false);
}

#ifdef HAVE_TDM
// 2D tile DMA: tile_rows x tile_cols f32 elements, row stride (elements) in
// global memory, packed contiguously into LDS at lds_off. D# per ISA 08 §8.3/8.4.
// This toolchain exposes the 6-arg builtin (extra int32x8 group, zero-filled).
__device__ __forceinline__ void tdm_load_2d(unsigned int lds_off,
                                            const float* gptr, int tile_cols,
                                            int tile_rows, int row_stride) {
  unsigned long long ga = (unsigned long long)(size_t)gptr;
  v4u g0;
  g0.x = 0x1u;                                   // count=1, is_restore=0
  g0.y = lds_off;                                // lds_addr (bytes)
  g0.z = (unsigned int)(ga & 0xFFFFFFFFu);       // global_addr[31:0]
  g0.w = (unsigned int)((ga >> 32) & 0x01FFFFFFu) | (2u << 30);  // [56:32] | type=2
  v8i g1;
  g1[0] = 0x00020000;                            // data_size=2 (4 bytes), mask=0
  g1[1] = (tile_cols & 0xFFFF) << 16;            // tensor_dim0[15:0]
  g1[2] = ((unsigned)tile_cols >> 16) | ((tile_rows & 0xFFFF) << 16); // d0 hi | d1 lo
  g1[3] = ((unsigned)tile_rows >> 16) | ((tile_cols & 0xFFFF) << 16); // d1 hi | tile_dim0
  g1[4] = (tile_rows & 0xFFFF);                  // tile_dim1 | tile_dim2=0
  g1[5] = row_stride;                            // tensor_dim0_stride[31:0]
  g1[6] = 0;                                     // stride0 hi | stride1 lo
  g1[7] = 0;
  v4i g2 = {0, 0, 0, 0};
  v4i g3 = {0, 0, 0, 0};
  v8i g4 = {0, 0, 0, 0, 0, 0, 0, 0};
  __builtin_amdgcn_tensor_load_to_lds(g0, g1, g2, g3, g4, 0);
}
#endif

// ---------------------------------------------------------------------------
// 1. objectness_label
// ---------------------------------------------------------------------------
__global__ __launch_bounds__(256) void label_kernel(
    const float* __restrict__ xyz, const float* __restrict__ gt,
    float* __restrict__ out_label) {
  __shared__ float g[KG_ * 3];
  const int b = blockIdx.x;
  const int k = threadIdx.x;
  if (threadIdx.x < KG_ * 3) g[threadIdx.x] = gt[b * KG_ * 3 + threadIdx.x];
  __syncthreads();
  const float x = xyz[(b * K_ + k) * 3 + 0];
  const float y = xyz[(b * K_ + k) * 3 + 1];
  const float z = xyz[(b * K_ + k) * 3 + 2];
  float mn = 3.4e38f;
  #pragma unroll 4
  for (int j = 0; j < KG_; ++j) {
    const float dx = x - g[3 * j + 0];
    const float dy = y - g[3 * j + 1];
    const float dz = z - g[3 * j + 2];
    mn = fminf(mn, dx * dx + dy * dy + dz * dz);
  }
  out_label[b * K_ + k] = (sqrtf(mn + 1e-6f) < 0.3f) ? 1.0f : 0.0f;
}

// ---------------------------------------------------------------------------
// 2. get_index: stable compaction via LDS prefix sum
// ---------------------------------------------------------------------------
__global__ __launch_bounds__(256) void index_kernel(
    const int* __restrict__ pred, float* __restrict__ idx_obj,
    float* __restrict__ sum1) {
  __shared__ int scan[256];
  __shared__ int sel[256];
  const int b = blockIdx.x;
  const int i = threadIdx.x;
  const int p = (pred[b * K_ + i] != 0) ? 1 : 0;
  scan[i] = p;
  __syncthreads();
  for (int off = 1; off < 256; off <<= 1) {
    const int v = scan[i];
    const int add = (i >= off) ? scan[i - off] : 0;
    __syncthreads();
    scan[i] = v + add;
    __syncthreads();
  }
  const int rank = scan[i] - p;
  if (p) sel[rank] = i;
  const int num = scan[255];
  __syncthreads();
  idx_obj[b * K_ + i] = (i < num) ? (float)sel[i] : (float)i;
  if (i == 0) sum1[b] = (float)num;
}

// ---------------------------------------------------------------------------
// 3. GEMM1: ws0[b,o,k] = W1 @ concat(features,rn) + b1
//    workgroup = (batch, 64-col K-slab); TDM stages 256x64 panel into LDS;
//    8 waves = 4 O-blocks x 2 K-subblocks, 2x2 WMMA register tiles each.
// ---------------------------------------------------------------------------
__global__ __launch_bounds__(256) void gemm1_kernel(
    const float* __restrict__ feat0, const float* __restrict__ feat1,
    const float* __restrict__ W1, const float* __restrict__ b1,
    float* __restrict__ out) {
  __shared__ float panel[C2_ * 64];  // 64 KB
  const int b = blockIdx.x >> 2;
  const int ks = blockIdx.x & 3;
  const int tid = threadIdx.x;
  const int lane = tid & 31;
  const int wave = tid >> 5;
#ifdef HAVE_TDM
  if (wave == 0) {
    const unsigned int lbase = (unsigned int)(size_t)(&panel[0]);
    tdm_load_2d(lbase, feat0 + (size_t)b * C_ * K_ + ks * 64, 64, C_, K_);
    tdm_load_2d(lbase + C_ * 64 * 4, feat1 + (size_t)b * C_ * K_ + ks * 64, 64,
                C_, K_);
    __builtin_amdgcn_s_wait_tensorcnt(0);
  }
#else
  for (int e = tid; e < C2_ * 64; e += 256) {
    const int r = e >> 6;
    const int c = e & 63;
    panel[e] = (r < C_) ? feat0[((size_t)b * C_ + r) * K_ + ks * 64 + c]
                        : feat1[((size_t)b * C_ + (r - C_)) * K_ + ks * 64 + c];
  }
#endif
  __syncthreads();

  const int ob = wave >> 1;    // O-block 0..3 (32 rows)
  const int ksub = wave & 1;   // K-sub 0..1 (32 cols)
  const int m = lane & 15;
  const int half = lane >> 4;
  const int col0 = ksub * 32 + m;
  const int col1 = col0 + 16;
  const int r0 = ob * 32 + m;
  const int r1 = r0 + 16;
  const float* Wr0 = W1 + r0 * C2_;
  const float* Wr1 = W1 + r1 * C2_;
  v8f acc00 = {}, acc01 = {}, acc10 = {}, acc11 = {};
  for (int c0 = 0; c0 < C2_; c0 += 4) {
    const int ca = c0 + 2 * half;
    const v2f a0 = *(const v2f*)(Wr0 + ca);
    const v2f a1 = *(const v2f*)(Wr1 + ca);
    v2f bv0, bv1;
    bv0.x = panel[ca * 64 + col0];
    bv0.y = panel[(ca + 1) * 64 + col0];
    bv1.x = panel[ca * 64 + col1];
    bv1.y = panel[(ca + 1) * 64 + col1];
    acc00 = wmma4(a0, bv0, acc00);
    acc01 = wmma4(a0, bv1, acc01);
    acc10 = wmma4(a1, bv0, acc10);
    acc11 = wmma4(a1, bv1, acc11);
  }
  const int n0 = ks * 64 + ksub * 32 + m;
  const int n1 = n0 + 16;
  #pragma unroll
  for (int v = 0; v < 8; ++v) {
    const int mg0 = ob * 32 + v + 8 * half;
    const int mg1 = mg0 + 16;
    const float bs0 = b1[mg0];
    const float bs1 = b1[mg1];
    out[((size_t)b * C_ + mg0) * K_ + n0] = acc00[v] + bs0;
    out[((size_t)b * C_ + mg0) * K_ + n1] = acc01[v] + bs0;
    out[((size_t)b * C_ + mg1) * K_ + n0] = acc10[v] + bs1;
    out[((size_t)b * C_ + mg1) * K_ + n1] = acc11[v] + bs1;
  }
}

// ---------------------------------------------------------------------------
// 4. BN stats (per channel, deterministic tree reduction -> fused scale/shift)
// ---------------------------------------------------------------------------
__global__ __launch_bounds__(256) void bnstats_kernel(
    const float* __restrict__ x, const float* __restrict__ g,
    const float* __restrict__ be, float* __restrict__ stats) {
  const int o = blockIdx.x;
  float s = 0.f, sq = 0.f;
  for (int i = threadIdx.x; i < B_ * K_; i += 256) {
    const int b = i >> 8;
    const int k = i & 255;
    const float v = x[((size_t)b * C_ + o) * K_ + k];
    s += v;
    sq += v * v;
  }
  __shared__ float ls[256], lq[256];
  ls[threadIdx.x] = s;
  lq[threadIdx.x] = sq;
  __syncthreads();
  for (int st = 128; st > 0; st >>= 1) {
    if (threadIdx.x < st) {
      ls[threadIdx.x] += ls[threadIdx.x + st];
      lq[threadIdx.x] += lq[threadIdx.x + st];
    }
    __syncthreads();
  }
  if (threadIdx.x == 0) {
    const float inv_n = 1.0f / (float)(B_ * K_);
    const float mean = ls[0] * inv_n;
    const float var = lq[0] * inv_n - mean * mean;
    const float scale = rsqrtf(var + 1e-5f) * g[o];
    stats[o * 2 + 0] = scale;
    stats[o * 2 + 1] = be[o] - mean * scale;
  }
}

// ---------------------------------------------------------------------------
// 5. GEMM2: TDM-stage raw ws0 panel, BN+ReLU in place in LDS, then WMMA
// ---------------------------------------------------------------------------
__global__ __launch_bounds__(256) void gemm2_kernel(
    const float* __restrict__ ws0, const float* __restrict__ stats,
    const float* __restrict__ W2, const float* __restrict__ b2,
    float* __restrict__ out) {
  __shared__ float panel[C_ * 64];  // 32 KB
  const int b = blockIdx.x >> 2;
  const int ks = blockIdx.x & 3;
  const int tid = threadIdx.x;
  const int lane = tid & 31;
  const int wave = tid >> 5;
#ifdef HAVE_TDM
  if (wave == 0) {
    tdm_load_2d((unsigned int)(size_t)(&panel[0]),
                ws0 + (size_t)b * C_ * K_ + ks * 64, 64, C_, K_);
    __builtin_amdgcn_s_wait_tensorcnt(0);
  }
#else
  for (int e = tid; e < C_ * 64; e += 256)
    panel[e] = ws0[((size_t)b * C_ + (e >> 6)) * K_ + ks * 64 + (e & 63)];
#endif
  __syncthreads();
  for (int e = tid; e < C_ * 64; e += 256) {  // BN+ReLU once per element
    const int r = e >> 6;
    panel[e] = fmaxf(panel[e] * stats[2 * r] + stats[2 * r + 1], 0.0f);
  }
  __syncthreads();

  const int ob = wave >> 1;
  const int ksub = wave & 1;
  const int m = lane & 15;
  const int half = lane >> 4;
  const int col0 = ksub * 32 + m;
  const int col1 = col0 + 16;
  const float* Wr0 = W2 + (ob * 32 + m) * C_;
  const float* Wr1 = Wr0 + 16 * C_;
  v8f acc00 = {}, acc01 = {}, acc10 = {}, acc11 = {};
  for (int c0 = 0; c0 < C_; c0 += 4) {
    const int ca = c0 + 2 * half;
    const v2f a0 = *(const v2f*)(Wr0 + ca);
    const v2f a1 = *(const v2f*)(Wr1 + ca);
    v2f bv0, bv1;
    bv0.x = panel[ca * 64 + col0];
    bv0.y = panel[(ca + 1) * 64 + col0];
    bv1.x = panel[ca * 64 + col1];
    bv1.y = panel[(ca + 1) * 64 + col1];
    acc00 = wmma4(a0, bv0, acc00);
    acc01 = wmma4(a0, bv1, acc01);
    acc10 = wmma4(a1, bv0, acc10);
    acc11 = wmma4(a1, bv1, acc11);
  }
  const int n0 = ks * 64 + ksub * 32 + m;
  const int n1 = n0 + 16;
  #pragma unroll
  for (int v = 0; v < 8; ++v) {
    const int mg0 = ob * 32 + v + 8 * half;
    const int mg1 = mg0 + 16;
    const float bs0 = b2[mg0];
    const float bs1 = b2[mg1];
    out[((size_t)b * C_ + mg0) * K_ + n0] = acc00[v] + bs0;
    out[((size_t)b * C_ + mg0) * K_ + n1] = acc01[v] + bs0;
    out[((size_t)b * C_ + mg1) * K_ + n0] = acc10[v] + bs1;
    out[((size_t)b * C_ + mg1) * K_ + n1] = acc11[v] + bs1;
  }
}

// ---------------------------------------------------------------------------
// 6. GEMM3 (117x128) + channel-scatter epilogue
// ---------------------------------------------------------------------------
__device__ __forceinline__ void scatter_out(int o, float t, float* orow,
                                            const float* xyz_row,
                                            const float* mean_size) {
  if (o < 3) {
    orow[o] = t + xyz_row[o];
  } else if (o < 15) {
    orow[o] = t;
  } else if (o < 27) {
    orow[o] = t;
    orow[o + 12] = t * PI_OVER_12;
  } else if (o < 45) {
    orow[o + 12] = t;
  } else if (o < 99) {
    orow[o + 12] = t;
    orow[o + 66] = t * mean_size[o - 45];
  } else {
    orow[o + 66] = t;
  }
}

__global__ __launch_bounds__(256) void gemm3_kernel(
    const float* __restrict__ ws1, const float* __restrict__ stats,
    const float* __restrict__ W3, const float* __restrict__ b3,
    const float* __restrict__ xyz, const float* __restrict__ mean_size,
    float* __restrict__ out) {
  __shared__ float panel[C_ * 64];  // 32 KB
  const int b = blockIdx.x >> 2;
  const int ks = blockIdx.x & 3;
  const int tid = threadIdx.x;
  const int lane = tid & 31;
  const int wave = tid >> 5;
#ifdef HAVE_TDM
  if (wave == 0) {
    tdm_load_2d((unsigned int)(size_t)(&panel[0]),
                ws1 + (size_t)b * C_ * K_ + ks * 64, 64, C_, K_);
    __builtin_amdgcn_s_wait_tensorcnt(0);
  }
#else
  for (int e = tid; e < C_ * 64; e += 256)
    panel[e] = ws1[((size_t)b * C_ + (e >> 6)) * K_ + ks * 64 + (e & 63)];
#endif
  __syncthreads();
  for (int e = tid; e < C_ * 64; e += 256) {
    const int r = e >> 6;
    panel[e] = fmaxf(panel[e] * stats[2 * r] + stats[2 * r + 1], 0.0f);
  }
  __syncthreads();

  const int ob = wave >> 1;
  const int ksub = wave & 1;
  const int m = lane & 15;
  const int half = lane >> 4;
  const int col0 = ksub * 32 + m;
  const int col1 = col0 + 16;
  const int r0 = ob * 32 + m;         // always < 117 (max 111)
  const int r1 = r0 + 16;             // up to 127: clamp + zero
  const bool v1 = (r1 < O3_);
  const int r1c = v1 ? r1 : 0;
  const float* Wr0 = W3 + r0 * C_;
  const float* Wr1 = W3 + r1c * C_;
  v8f acc00 = {}, acc01 = {}, acc10 = {}, acc11 = {};
  for (int c0 = 0; c0 < C_; c0 += 4) {
    const int ca = c0 + 2 * half;
    const v2f a0 = *(const v2f*)(Wr0 + ca);
    v2f a1 = *(const v2f*)(Wr1 + ca);
    a1.x = v1 ? a1.x : 0.0f;
    a1.y = v1 ? a1.y : 0.0f;
    v2f bv0, bv1;
    bv0.x = panel[ca * 64 + col0];
    bv0.y = panel[(ca + 1) * 64 + col0];
    bv1.x = panel[ca * 64 + col1];
    bv1.y = panel[(ca + 1) * 64 + col1];
    acc00 = wmma4(a0, bv0, acc00);
    acc01 = wmma4(a0, bv1, acc01);
    acc10 = wmma4(a1, bv0, acc10);
    acc11 = wmma4(a1, bv1, acc11);
  }
  const int n0 = ks * 64 + ksub * 32 + m;
  const int n1 = n0 + 16;
  float* orow0 = out + (size_t)(b * K_ + n0) * OUTF_;
  float* orow1 = out + (size_t)(b * K_ + n1) * OUTF_;
  const float* xr0 = xyz + (size_t)(b * K_ + n0) * 3;
  const float* xr1 = xyz + (size_t)(b * K_ + n1) * 3;
  #pragma unroll
  for (int v = 0; v < 8; ++v) {
    const int o0 = ob * 32 + v + 8 * half;
    const int o1 = o0 + 16;
    if (o0 < O3_) {
      const float bo = b3[o0];
      scatter_out(o0, acc00[v] + bo, orow0, xr0, mean_size);
      scatter_out(o0, acc01[v] + bo, orow1, xr1, mean_size);
    }
    if (o1 < O3_) {
      const float bo = b3[o1];
      scatter_out(o1, acc10[v] + bo, orow0, xr0, mean_size);
      scatter_out(o1, acc11[v] + bo, orow1, xr1, mean_size);
    }
  }
}

// ---------------------------------------------------------------------------
extern "C" void kernel_launch(void* const* d_in, const int* in_sizes, int n_in,
                              void* d_out, int out_size, void* d_ws, size_t ws_size,
                              hipStream_t stream) {
  (void)in_sizes; (void)n_in; (void)out_size; (void)ws_size;
  const float* xyz       = (const float*)d_in[0];
  const float* features  = (const float*)d_in[1];
  const float* rn        = (const float*)d_in[2];
  const float* gt_center = (const float*)d_in[3];
  const int*   pred      = (const int*)d_in[4];
  const float* W1 = (const float*)d_in[5];
  const float* b1 = (const float*)d_in[6];
  const float* g1 = (const float*)d_in[7];
  const float* be1 = (const float*)d_in[8];
  const float* W2 = (const float*)d_in[9];
  const float* b2 = (const float*)d_in[10];
  const float* g2 = (const float*)d_in[11];
  const float* be2 = (const float*)d_in[12];
  const float* W3 = (const float*)d_in[13];
  const float* b3 = (const float*)d_in[14];
  const float* mean_size = (const float*)d_in[15];

  float* ws0 = (float*)d_ws;                         // B*C*K
  float* ws1 = ws0 + (size_t)B_ * C_ * K_;           // B*C*K
  float* stats1 = ws1 + (size_t)B_ * C_ * K_;        // 2*C
  float* stats2 = stats1 + 2 * C_;                   // 2*C

  float* out_main = (float*)d_out;                   // B*K*183
  float* out_idx  = out_main + (size_t)B_ * K_ * OUTF_;
  float* out_sum  = out_idx + (size_t)B_ * K_;
  float* out_lab  = out_sum + B_;

  label_kernel<<<B_, 256, 0, stream>>>(xyz, gt_center, out_lab);
  index_kernel<<<B_, 256, 0, stream>>>(pred, out_idx, out_sum);

  const int nblk = B_ * (K_ / 64);  // 1024 workgroups, 8 waves each
  gemm1_kernel<<<nblk, 256, 0, stream>>>(features, rn, W1, b1, ws0);
  bnstats_kernel<<<C_, 256, 0, stream>>>(ws0, g1, be1, stats1);
  gemm2_kernel<<<nblk, 256, 0, stream>>>(ws0, stats1, W2, b2, ws1);
  bnstats_kernel<<<C_, 256, 0, stream>>>(ws1, g2, be2, stats2);
  gemm3_kernel<<<nblk, 256, 0, stream>>>(ws1, stats2, W3, b3, xyz, mean_size,
                                         out_main);
}